// UltimateRA_79929341378600
// MI455X (gfx1250) — compile-verified
//
#include <hip/hip_runtime.h>
#include <math.h>

// ---------------------------------------------------------------------------
// UltimateRA fused attention for MI455X (gfx1250).
//   * all matmuls on v_wmma_f32_16x16x4_f32 (exact fp32, matrix pipe)
//   * GEMMs stage tiles in LDS via global_load_async_to_lds_b128 (ASYNCcnt),
//     double-buffered so DMA overlaps WMMA compute
//   * causal flash attention with per-wave LDS softmax (s_wait_dscnt)
// ---------------------------------------------------------------------------

typedef float v2f __attribute__((ext_vector_type(2)));
typedef float v8f __attribute__((ext_vector_type(8)));

constexpr int NH   = 12;    // heads
constexpr int SEQ  = 2048;  // tokens
constexpr int CDIM = 768;   // channels
constexpr int HD   = 64;    // head dim

static __device__ __forceinline__ v8f wmma4(v2f a, v2f b, v8f c) {
  // D = A(16x4,f32) * B(4x16,f32) + C(16x16,f32)
  return __builtin_amdgcn_wmma_f32_16x16x4_f32(false, a, false, b, (short)0, c,
                                               false, false);
}

// low 32 bits of a generic pointer into __shared__ == LDS byte offset
static __device__ __forceinline__ unsigned lds_addr(const void* p) {
  return (unsigned)(unsigned long long)p;
}

// one 16-byte async copy global -> LDS (tracked by ASYNCcnt)
static __device__ __forceinline__ void async_cp16(const float* g, const float* l) {
  asm volatile("global_load_async_to_lds_b128 %0, %1, off"
               :: "v"(lds_addr(l)), "v"(g)
               : "memory");
}

static __device__ __forceinline__ void wait_async0() {
  asm volatile("s_wait_asynccnt 0x0" ::: "memory");
}

// ---------------------------------------------------------------------------
// C[M,N] = A[M,K] @ W[N,K]^T
// Block (8 waves) -> 128x64 C tile. K-blocks of 32 staged in LDS via async
// b128 copies, double-buffered. Wave w computes rows w*16..w*16+15; the B
// panel in LDS is shared by all 8 waves. Requires M%128==0, N%64==0, K%32==0.
// ---------------------------------------------------------------------------
constexpr int KB   = 32;  // K-block
constexpr int LSTR = 36;  // padded LDS row stride (144B: 16B-aligned, 16 banks)

__global__ void __launch_bounds__(256) ra_gemm_xwt(
    const float* __restrict__ A, const float* __restrict__ W,
    float* __restrict__ C, int M, int N, int K) {
  __shared__ __align__(16) float sA[2][128 * LSTR];
  __shared__ __align__(16) float sB[2][64 * LSTR];

  const int tid  = threadIdx.x;
  const int wib  = tid >> 5;
  const int lane = tid & 31;
  const int r    = lane & 15;
  const int hh   = lane >> 4;

  const int ntiles = N >> 6;
  const int mb = blockIdx.x / ntiles;
  const int nb = blockIdx.x % ntiles;
  const int m0 = mb * 128;
  const int n0 = nb * 64;

  auto issue = [&](int buf, int kc) {
    // A panel: 128 rows x 32 floats = 1024 x 16B, 4 copies per thread
#pragma unroll
    for (int i = 0; i < 4; ++i) {
      const int e = tid + 256 * i;
      const int row = e >> 3, seg = e & 7;
      async_cp16(A + (size_t)(m0 + row) * K + kc + seg * 4,
                 &sA[buf][row * LSTR + seg * 4]);
    }
    // B panel: 64 rows x 32 floats = 512 x 16B, 2 copies per thread
#pragma unroll
    for (int i = 0; i < 2; ++i) {
      const int e = tid + 256 * i;
      const int row = e >> 3, seg = e & 7;
      async_cp16(W + (size_t)(n0 + row) * K + kc + seg * 4,
                 &sB[buf][row * LSTR + seg * 4]);
    }
  };

  v8f a0 = {}, a1 = {}, a2 = {}, a3 = {};
  const int nkb = K / KB;

  issue(0, 0);
  for (int kb = 0; kb < nkb; ++kb) {
    wait_async0();       // own async group for buf[kb&1] complete
    __syncthreads();     // all waves' copies visible / prior reads finished
    if (kb + 1 < nkb) issue((kb + 1) & 1, (kb + 1) * KB);  // overlap next DMA

    const float* pa = &sA[kb & 1][(wib * 16 + r) * LSTR + 2 * hh];
    const float* pb = &sB[kb & 1][r * LSTR + 2 * hh];
#pragma unroll
    for (int c = 0; c < 8; ++c) {
      v2f av = *(const v2f*)(pa + c * 4);
      a0 = wmma4(av, *(const v2f*)(pb + 0 * 16 * LSTR + c * 4), a0);
      a1 = wmma4(av, *(const v2f*)(pb + 1 * 16 * LSTR + c * 4), a1);
      a2 = wmma4(av, *(const v2f*)(pb + 2 * 16 * LSTR + c * 4), a2);
      a3 = wmma4(av, *(const v2f*)(pb + 3 * 16 * LSTR + c * 4), a3);
    }
  }

#pragma unroll
  for (int i = 0; i < 8; ++i) {
    const int m = m0 + wib * 16 + i + 8 * hh;  // C layout: m = vgpr + 8*half
    float* crow = C + (size_t)m * N + n0 + r;
    crow[0]  = a0[i];
    crow[16] = a1[i];
    crow[32] = a2[i];
    crow[48] = a3[i];
  }
}

// ---------------------------------------------------------------------------
// db[h,t] = w_disc[h] * (d_bias[h,t] - mean_t d_bias[h,:])
// ---------------------------------------------------------------------------
__global__ void __launch_bounds__(256) ra_dbias(
    const float* __restrict__ d_bias, const float* __restrict__ w_disc,
    float* __restrict__ db) {
  const int h = blockIdx.x;
  __shared__ float red[256];
  float s = 0.f;
  for (int t = threadIdx.x; t < SEQ; t += 256) s += d_bias[h * SEQ + t];
  red[threadIdx.x] = s;
  __syncthreads();
  for (int k = 128; k > 0; k >>= 1) {
    if (threadIdx.x < k) red[threadIdx.x] += red[threadIdx.x + k];
    __syncthreads();
  }
  const float mean = red[0] * (1.0f / SEQ);
  const float wd   = w_disc[h];
  for (int t = threadIdx.x; t < SEQ; t += 256)
    db[h * SEQ + t] = wd * (d_bias[h * SEQ + t] - mean);
}

// ---------------------------------------------------------------------------
// Augmented features (folds w_std / reciprocal routing / 0.125 into Qf,Kf):
//   Qf[h,t,0:56]=0.125*sqrt(w_std)*q ; tail = 0.125*sqrt(w_rec)*(k@W_recip) if RA
//   Kf[h,t,0:56]=      sqrt(w_std)*k ; tail =       sqrt(w_rec)*(q@W_recip) if RA
//   baseline heads: full 64 dims of scaled q/k.  Vt[h,d,t] = v.
// ---------------------------------------------------------------------------
__global__ void __launch_bounds__(256) ra_feat(
    const float* __restrict__ qkv, const float* __restrict__ W_recip,
    const float* __restrict__ w_std, const float* __restrict__ w_rec,
    float* __restrict__ Qf, float* __restrict__ Kf, float* __restrict__ Vt) {
  const int idx = blockIdx.x * blockDim.x + threadIdx.x;   // < NH*SEQ exactly
  const int h = idx / SEQ;
  const int t = idx % SEQ;

  const float* q = qkv + (size_t)t * (3 * CDIM) + h * HD;
  const float* k = q + CDIM;
  const float* v = q + 2 * CDIM;

  const float wsv = sqrtf(fmaxf(w_std[h], 1e-8f));
  const float wrv = fmaxf(w_rec[h], 1e-8f);
  const bool  ra  = w_rec[h] > 0.1f;

  float* qdst = Qf + ((size_t)h * SEQ + t) * HD;
  float* kdst = Kf + ((size_t)h * SEQ + t) * HD;
  float* vdst = Vt + (size_t)h * HD * SEQ + t;

  const float qsc = 0.125f * wsv;  // fold 1/sqrt(D) into query side only
  float accQ[8] = {0, 0, 0, 0, 0, 0, 0, 0};
  float accK[8] = {0, 0, 0, 0, 0, 0, 0, 0};

  for (int d = 0; d < HD; ++d) {
    const float qd = q[d], kd = k[d];
    vdst[(size_t)d * SEQ] = v[d];
    if (d < 56 || !ra) { qdst[d] = qsc * qd; kdst[d] = wsv * kd; }
#pragma unroll
    for (int rr = 0; rr < 8; ++rr) {
      const float w = W_recip[d * 8 + rr];   // uniform -> scalar loads
      accQ[rr] += qd * w;
      accK[rr] += kd * w;
    }
  }
  if (ra) {
    const float swr = sqrtf(wrv);
#pragma unroll
    for (int rr = 0; rr < 8; ++rr) {
      qdst[56 + rr] = 0.125f * swr * accK[rr];  // reciprocal: query carries KW
      kdst[56 + rr] = swr * accQ[rr];           // key carries QW
    }
  }
}

// ---------------------------------------------------------------------------
// Causal flash attention. One wave owns a 16-query tile of one head; online
// softmax state lives in lanes 0..15; S/P tiles bounce through a private LDS
// slab (wave-internal, so only s_wait_dscnt -- no block barrier).
// ---------------------------------------------------------------------------
__global__ void __launch_bounds__(256) ra_flash(
    const float* __restrict__ Qf, const float* __restrict__ Kf,
    const float* __restrict__ Vt, const float* __restrict__ db,
    float* __restrict__ Out) {
  __shared__ __align__(16) float lds[8 * 288];
  const int wib  = threadIdx.x >> 5;
  const int wave = blockIdx.x * 8 + wib;
  const int lane = threadIdx.x & 31;
  const int r  = lane & 15;
  const int hh = lane >> 4;
  const int h  = wave >> 7;       // SEQ/16 = 128 q-tiles per head
  const int qt = wave & 127;
  float* sl = lds + wib * 288;    // [0..255]=S/P, [256..271]=alpha, [272..287]=1/l

  const float* qbase = Qf + ((size_t)(h * SEQ + qt * 16 + r)) * HD + 2 * hh;
  v2f qa[16];
#pragma unroll
  for (int c = 0; c < 16; ++c) qa[c] = *(const v2f*)(qbase + 4 * c);

  v8f o0 = {}, o1 = {}, o2 = {}, o3 = {};
  float mrow = -__builtin_inff();   // valid in lanes 0..15
  float lrow = 0.f;

  const float* kfh = Kf + (size_t)h * SEQ * HD;
  const float* vth = Vt + (size_t)h * HD * SEQ;
  const float* dbh = db + h * SEQ;

  for (int kb = 0; kb <= qt; ++kb) {
    const float* kbase = kfh + (size_t)(kb * 16 + r) * HD + 2 * hh;
    if (kb < qt)  // warm next K tile (global_prefetch_b8)
      __builtin_prefetch(kfh + (size_t)((kb + 1) * 16 + r) * HD, 0, 1);

    // S = Qf * Kf^T  (16 WMMAs, K=64)
    v8f s = {};
#pragma unroll
    for (int c = 0; c < 16; ++c)
      s = wmma4(qa[c], *(const v2f*)(kbase + 4 * c), s);

    // add discoverability bias (per key j), causal-mask diagonal block
    const float bias = dbh[kb * 16 + r];
    const bool  diag = (kb == qt);
    const int   jg   = kb * 16 + r;
#pragma unroll
    for (int i = 0; i < 8; ++i) {
      const int m = i + 8 * hh;
      float val = s[i] + bias;
      if (diag && jg > qt * 16 + m) val = -__builtin_inff();
      sl[m * 16 + r] = val;
    }
    asm volatile("s_wait_dscnt 0x0" ::: "memory");

    // online softmax, row m handled by lane m
    if (lane < 16) {
      float mx = mrow;
#pragma unroll
      for (int jj = 0; jj < 16; ++jj) mx = fmaxf(mx, sl[lane * 16 + jj]);
      const float alpha = expf(mrow - mx);
      float sum = 0.f;
#pragma unroll
      for (int jj = 0; jj < 16; ++jj) {
        const float p = expf(sl[lane * 16 + jj] - mx);
        sl[lane * 16 + jj] = p;
        sum += p;
      }
      lrow = lrow * alpha + sum;
      mrow = mx;
      sl[256 + lane] = alpha;
    }
    asm volatile("s_wait_dscnt 0x0" ::: "memory");

    // rescale O accumulators by alpha[row]
#pragma unroll
    for (int i = 0; i < 8; ++i) {
      const float al = sl[256 + i + 8 * hh];
      o0[i] *= al; o1[i] *= al; o2[i] *= al; o3[i] *= al;
    }

    // O += P * V   (A = P chunk from LDS, B = Vt rows, 16 WMMAs)
#pragma unroll
    for (int c = 0; c < 4; ++c) {
      v2f pa = *(const v2f*)(sl + r * 16 + 4 * c + 2 * hh);
      const int kcol = kb * 16 + 4 * c + 2 * hh;
      o0 = wmma4(pa, *(const v2f*)(vth + (size_t)(0 + r) * SEQ + kcol), o0);
      o1 = wmma4(pa, *(const v2f*)(vth + (size_t)(16 + r) * SEQ + kcol), o1);
      o2 = wmma4(pa, *(const v2f*)(vth + (size_t)(32 + r) * SEQ + kcol), o2);
      o3 = wmma4(pa, *(const v2f*)(vth + (size_t)(48 + r) * SEQ + kcol), o3);
    }
  }

  if (lane < 16) sl[272 + lane] = 1.0f / lrow;
  asm volatile("s_wait_dscnt 0x0" ::: "memory");

  // write [T][C] layout so the projection GEMM streams contiguous rows
#pragma unroll
  for (int i = 0; i < 8; ++i) {
    const float linv = sl[272 + i + 8 * hh];
    const int t = qt * 16 + i + 8 * hh;
    float* orow = Out + (size_t)t * CDIM + h * HD + r;
    orow[0]  = o0[i] * linv;
    orow[16] = o1[i] * linv;
    orow[32] = o2[i] * linv;
    orow[48] = o3[i] * linv;
  }
}

// ---------------------------------------------------------------------------
extern "C" void kernel_launch(void* const* d_in, const int* in_sizes, int n_in,
                              void* d_out, int out_size, void* d_ws,
                              size_t ws_size, hipStream_t stream) {
  const float* x       = (const float*)d_in[0];
  const float* W_attn  = (const float*)d_in[1];
  const float* W_proj  = (const float*)d_in[2];
  const float* W_recip = (const float*)d_in[3];
  const float* w_std   = (const float*)d_in[4];
  const float* w_rec   = (const float*)d_in[5];
  const float* w_disc  = (const float*)d_in[6];
  const float* d_bias  = (const float*)d_in[7];
  float* out = (float*)d_out;

  float* ws  = (float*)d_ws;
  float* qkv = ws;                                   // SEQ * 3C
  float* Qf  = qkv + (size_t)SEQ * 3 * CDIM;         // NH*SEQ*HD
  float* Kf  = Qf + (size_t)NH * SEQ * HD;
  float* Vt  = Kf + (size_t)NH * SEQ * HD;
  float* db  = Vt + (size_t)NH * SEQ * HD;           // NH*SEQ
  float* att = db + (size_t)NH * SEQ;                // SEQ*C
  (void)in_sizes; (void)n_in; (void)out_size; (void)ws_size;

  // 1) qkv = x @ W_attn^T : M=2048, N=2304, K=768 -> 16*36 = 576 blocks
  {
    const int M = SEQ, N = 3 * CDIM, K = CDIM;
    ra_gemm_xwt<<<(M / 128) * (N / 64), 256, 0, stream>>>(x, W_attn, qkv,
                                                          M, N, K);
  }
  // 2) centered discoverability bias
  ra_dbias<<<NH, 256, 0, stream>>>(d_bias, w_disc, db);
  // 3) augmented features + V transpose
  ra_feat<<<(NH * SEQ) / 256, 256, 0, stream>>>(qkv, W_recip, w_std, w_rec,
                                                Qf, Kf, Vt);
  // 4) flash attention: 12 heads * 128 q-tiles = 1536 waves
  ra_flash<<<(NH * (SEQ / 16)) / 8, 256, 0, stream>>>(Qf, Kf, Vt, db, att);
  // 5) out = att @ W_proj^T : M=2048, N=768, K=768 -> 16*12 = 192 blocks
  {
    const int M = SEQ, N = CDIM, K = CDIM;
    ra_gemm_xwt<<<(M / 128) * (N / 64), 256, 0, stream>>>(att, W_proj, out,
                                                          M, N, K);
  }
}